// TemporalGNN_51316269253248
// MI455X (gfx1250) — compile-verified
//
#include <hip/hip_runtime.h>
#include <cmath>

#define NN 50000
#define CC 128
#define EE 600000
#define TT 5            // M-tiles (of 16 rows) per block in the WMMA GEMM

typedef __attribute__((ext_vector_type(2))) float v2f;
typedef __attribute__((ext_vector_type(8))) float v8f;
typedef __attribute__((ext_vector_type(4))) int   v4i;

#if defined(__HIP_DEVICE_COMPILE__) && \
    __has_builtin(__builtin_amdgcn_global_load_async_to_lds_b128) && \
    __has_builtin(__builtin_amdgcn_s_wait_asynccnt)
#define HAVE_ASYNC_LDS 1
#else
#define HAVE_ASYNC_LDS 0
#if defined(__HIP_DEVICE_COMPILE__)
#warning "gfx1250 async-to-LDS builtins not available; using synchronous LDS staging"
#endif
#endif

__device__ __forceinline__ float waveRedSum(float v) {
#pragma unroll
    for (int o = 16; o > 0; o >>= 1) v += __shfl_down(v, o, 32);
    return v;
}

__device__ __forceinline__ unsigned orderKey(float f) {
    unsigned u = __float_as_uint(f);
    return (u & 0x80000000u) ? ~u : (u | 0x80000000u);
}

// ---------------- 1) 1/||p|| ----------------
__global__ void pnorm_kernel(const float* __restrict__ p, float* __restrict__ inv_norm) {
    int lane = threadIdx.x;
    float s = 0.f;
#pragma unroll
    for (int j = 0; j < 4; ++j) { float t = p[lane + 32 * j]; s += t * t; }
    s = waveRedSum(s);
    if (lane == 0) *inv_norm = rsqrtf(s);
}

// ---------------- 2) score = (x @ p) * inv_norm ; orderable key ----------------
__global__ void score_kernel(const float* __restrict__ x, const float* __restrict__ p,
                             const float* __restrict__ inv_norm,
                             float* __restrict__ score_f, unsigned* __restrict__ score_u) {
    __shared__ float ps[CC];
    int tid = threadIdx.x;
    if (tid < CC) ps[tid] = p[tid];
    __syncthreads();
    int wave = tid >> 5, lane = tid & 31;
    int row = blockIdx.x * 8 + wave;           // 6250 * 8 == 50000 exactly
    const float4* xr = (const float4*)(x + (size_t)row * CC);
    float4 a = xr[lane];
    float s = a.x * ps[lane * 4] + a.y * ps[lane * 4 + 1] +
              a.z * ps[lane * 4 + 2] + a.w * ps[lane * 4 + 3];
    s = waveRedSum(s);
    if (lane == 0) {
        s *= *inv_norm;
        score_f[row] = s;
        score_u[row] = orderKey(s);
    }
}

// ---------------- 3) radix-select top-128 (exact threshold) ----------------
__global__ void rs_init(unsigned* __restrict__ hist, unsigned* __restrict__ st) {
    hist[threadIdx.x] = 0u;
    if (threadIdx.x == 0) { st[0] = 0u; st[1] = 128u; st[2] = 0u; }
}

__global__ void rs_hist(const unsigned* __restrict__ keys, unsigned* __restrict__ hist,
                        const unsigned* __restrict__ st, int pass) {
    __shared__ unsigned sh[256];
    int tid = threadIdx.x;
    sh[tid] = 0u;
    __syncthreads();
    unsigned prefix = st[0];
    unsigned pmask = pass ? (0xFFFFFFFFu << (32 - 8 * pass)) : 0u;
    int shift = 24 - 8 * pass;
    for (int i = blockIdx.x * blockDim.x + tid; i < NN; i += gridDim.x * blockDim.x) {
        unsigned u = keys[i];
        if ((u & pmask) == prefix) atomicAdd(&sh[(u >> shift) & 255u], 1u);
    }
    __syncthreads();
    if (sh[tid]) atomicAdd(&hist[tid], sh[tid]);
}

__global__ void rs_scan(unsigned* __restrict__ hist, unsigned* __restrict__ st, int pass) {
    unsigned krem = st[1];
    unsigned cum = 0u, digit = 0u;
    int shift = 24 - 8 * pass;
    for (int b = 255; b >= 0; --b) {
        unsigned c = hist[b];
        if (cum + c >= krem) { digit = (unsigned)b; break; }
        cum += c;
    }
    st[0] |= digit << shift;
    st[1] = krem - cum;
    for (int b = 0; b < 256; ++b) hist[b] = 0u;
}

__global__ void rs_collect(const unsigned* __restrict__ keys, unsigned* __restrict__ st,
                           unsigned* __restrict__ cand_idx, unsigned* __restrict__ cand_key) {
    unsigned thr = st[0];
    for (int i = blockIdx.x * blockDim.x + threadIdx.x; i < NN; i += gridDim.x * blockDim.x) {
        unsigned u = keys[i];
        if (u >= thr) {
            unsigned p = atomicAdd(&st[2], 1u);
            if (p < 1024u) { cand_idx[p] = (unsigned)i; cand_key[p] = u; }
        }
    }
}

// rank sort (value desc, index asc) -> matches jax.lax.top_k ordering
__global__ void rs_select(const unsigned* __restrict__ cand_idx, const unsigned* __restrict__ cand_key,
                          const unsigned* __restrict__ st, const float* __restrict__ score_f,
                          unsigned* __restrict__ perm, float* __restrict__ vals) {
    __shared__ unsigned sk[1024], si[1024];
    int tid = threadIdx.x;
    unsigned cnt = st[2]; if (cnt > 1024u) cnt = 1024u;
    for (unsigned m = tid; m < cnt; m += 128u) { sk[m] = cand_key[m]; si[m] = cand_idx[m]; }
    __syncthreads();
    for (unsigned j = tid; j < cnt; j += 128u) {
        unsigned uj = sk[j], ij = si[j];
        int rank = 0;
        for (unsigned m = 0; m < cnt; ++m) {
            unsigned um = sk[m], im = si[m];
            rank += (um > uj) || (um == uj && im < ij);
        }
        if (rank < 128) {
            perm[rank] = ij;
            vals[rank] = tanhf(score_f[ij]);
        }
    }
}

// ---------------- 4) x_tilde = x[perm] * vals ----------------
__global__ void build_xt(const float* __restrict__ x, const unsigned* __restrict__ perm,
                         const float* __restrict__ vals, float* __restrict__ xt) {
    int r = blockIdx.x;
    unsigned p = perm[r];
    float v = vals[r];
    xt[r * CC + threadIdx.x] = x[(size_t)p * CC + threadIdx.x] * v;
}

// ---------------- 5) GRU step -> evolved weight W [128,128] ----------------
__global__ void gru_kernel(const float* __restrict__ xt, const float* __restrict__ W0,
                           const float* __restrict__ w_ih, const float* __restrict__ w_hh,
                           const float* __restrict__ b_ih, const float* __restrict__ b_hh,
                           float* __restrict__ Wm) {
    __shared__ float xr[CC], hr[CC], gi[3 * CC], gh[3 * CC];
    int r = blockIdx.x, j = threadIdx.x;
    if (j < CC) { xr[j] = xt[r * CC + j]; hr[j] = W0[r * CC + j]; }
    __syncthreads();
    float a = b_ih[j], b = b_hh[j];
    const float* wi = w_ih + (size_t)j * CC;
    const float* wh = w_hh + (size_t)j * CC;
#pragma unroll 4
    for (int k = 0; k < CC; ++k) { a += xr[k] * wi[k]; b += hr[k] * wh[k]; }
    gi[j] = a; gh[j] = b;
    __syncthreads();
    if (j < CC) {
        float rg = 1.f / (1.f + expf(-(gi[j] + gh[j])));
        float z  = 1.f / (1.f + expf(-(gi[j + CC] + gh[j + CC])));
        float ng = tanhf(gi[j + 2 * CC] + rg * gh[j + 2 * CC]);
        Wm[r * CC + j] = (1.f - z) * ng + z * hr[j];
    }
}

// ---------------- 6) xw = x @ W  via V_WMMA_F32_16X16X4_F32 ----------------
// block = 256 thr = 8 waves; block owns TT tiles of 16 rows; wave w owns cols [16w,16w+16).
// x tiles double-buffered in LDS via async global->LDS copies (fallback: sync staging).
// B fragments (all of W needed by this wave) hoisted to registers once per block.

#if HAVE_ASYNC_LDS
__device__ __forceinline__ void async_copy_tile(const float* __restrict__ x, int row0,
                                                float* lds_dst, int tid) {
    // 16 rows x 128 floats, 512 b128 chunks, 2 per thread -> 2 async instrs per wave
#pragma unroll
    for (int h = 0; h < 2; ++h) {
        int idx = h * 256 + tid;
        int r = idx >> 5, c16 = idx & 31;
        const float* g = x + (size_t)(row0 + r) * CC + c16 * 4;
        float* l = lds_dst + r * 132 + c16 * 4;
        __attribute__((address_space(1))) void* gv =
            (__attribute__((address_space(1))) void*)(void*)g;
        __attribute__((address_space(3))) void* lv =
            (__attribute__((address_space(3))) void*)(void*)l;
        __builtin_amdgcn_global_load_async_to_lds_b128(
            (__attribute__((address_space(1))) v4i*)gv,
            (__attribute__((address_space(3))) v4i*)lv, 0, 0);
    }
}
#endif

__device__ __forceinline__ void sync_copy_tile(const float* __restrict__ x, int row0,
                                               float* lds_dst, int tid) {
#pragma unroll
    for (int h = 0; h < 2; ++h) {
        int idx = h * 256 + tid;
        int r = idx >> 5, c16 = idx & 31;
        const float4 v = *(const float4*)(x + (size_t)(row0 + r) * CC + c16 * 4);
        *(float4*)(lds_dst + r * 132 + c16 * 4) = v;
    }
}

__global__ void gemm_xw(const float* __restrict__ x, const float* __restrict__ Wm,
                        float* __restrict__ xw) {
    __shared__ float xs[2][16 * 132];                 // padded stride kills LDS bank conflicts
    int tid = threadIdx.x;
    int wave = tid >> 5, lane = tid & 31;
    int lm = lane & 15, lh = lane >> 4;
    int col0 = wave * 16;
    int base = blockIdx.x * (TT * 16);                // 625 * 80 == 50000 exactly

    // hoist all B fragments: bf[s][v] = W[4s + v + 2*lh][col0 + lm]
    v2f bf[32];
#pragma unroll
    for (int s = 0; s < 32; ++s) {
        bf[s][0] = Wm[(size_t)(4 * s + 2 * lh + 0) * CC + col0 + lm];
        bf[s][1] = Wm[(size_t)(4 * s + 2 * lh + 1) * CC + col0 + lm];
    }

#if HAVE_ASYNC_LDS
    async_copy_tile(x, base + 0 * 16, xs[0], tid);
    async_copy_tile(x, base + 1 * 16, xs[1], tid);
#endif

    for (int t = 0; t < TT; ++t) {
#if HAVE_ASYNC_LDS
        // async loads complete in order; <=2 outstanding means tile t is resident
        if (t + 1 < TT) __builtin_amdgcn_s_wait_asynccnt(2);
        else            __builtin_amdgcn_s_wait_asynccnt(0);
        __syncthreads();
#else
        sync_copy_tile(x, base + t * 16, xs[t & 1], tid);
        __syncthreads();
#endif
        const float* xsb = xs[t & 1];
        v8f c = {};
#pragma unroll
        for (int s = 0; s < 32; ++s) {
            v2f a;                                    // A[lm][4s + 2*lh + j]
            a[0] = xsb[lm * 132 + 4 * s + 2 * lh + 0];
            a[1] = xsb[lm * 132 + 4 * s + 2 * lh + 1];
            c = __builtin_amdgcn_wmma_f32_16x16x4_f32(false, a, false, bf[s],
                                                      (short)0, c, false, false);
        }
        int row0 = base + t * 16;
#pragma unroll
        for (int v = 0; v < 8; ++v)                   // D: VGPR v -> row v + 8*lh
            xw[(size_t)(row0 + v + 8 * lh) * CC + col0 + lm] = c[v];
        __syncthreads();                              // everyone done reading xs[t&1]
#if HAVE_ASYNC_LDS
        if (t + 2 < TT) async_copy_tile(x, base + (t + 2) * 16, xs[t & 1], tid);
#endif
    }
}

// ---------------- 7) misc elementwise ----------------
__global__ void zero_kernel(float* __restrict__ p, int n) {
    for (int i = blockIdx.x * blockDim.x + threadIdx.x; i < n; i += gridDim.x * blockDim.x)
        p[i] = 0.f;
}

__global__ void deg_kernel(const int* __restrict__ cols, const float* __restrict__ ew,
                           float* __restrict__ deg) {
    for (int i = blockIdx.x * blockDim.x + threadIdx.x; i < EE + NN; i += gridDim.x * blockDim.x) {
        if (i < EE) unsafeAtomicAdd(&deg[cols[i]], ew[i]);
        else        unsafeAtomicAdd(&deg[i - EE], 1.0f);
    }
}

__global__ void rsqrt_kernel(float* __restrict__ d) {
    int i = blockIdx.x * blockDim.x + threadIdx.x;
    if (i < NN) {
        float v = d[i];
        d[i] = v > 0.f ? rsqrtf(v) : 0.f;
    }
}

// ---------------- 8) edge scatter: acc[col] += norm * xw[row] ----------------
__global__ void scatter_kernel(const int* __restrict__ rows, const int* __restrict__ cols,
                               const float* __restrict__ ew, const float* __restrict__ dis,
                               const float* __restrict__ xw, float* __restrict__ acc) {
    int e = blockIdx.x * 8 + (threadIdx.x >> 5);
    if (e >= EE + NN) return;
    int r, c; float w;
    if (e < EE) { r = rows[e]; c = cols[e]; w = ew[e]; }
    else        { r = e - EE; c = r; w = 1.f; }
    float nrm = dis[r] * w * dis[c];
    int lane = threadIdx.x & 31;
    float4 v = ((const float4*)(xw + (size_t)r * CC))[lane];
    float* dst = acc + (size_t)c * CC + lane * 4;
    unsafeAtomicAdd(dst + 0, nrm * v.x);
    unsafeAtomicAdd(dst + 1, nrm * v.y);
    unsafeAtomicAdd(dst + 2, nrm * v.z);
    unsafeAtomicAdd(dst + 3, nrm * v.w);
}

// ---------------- 9) out = relu(acc) @ w_lin.T + b_lin ----------------
__global__ void head_kernel(const float* __restrict__ acc, const float* __restrict__ wlin,
                            const float* __restrict__ blin, float* __restrict__ out) {
    __shared__ float wl[CC];
    int tid = threadIdx.x;
    if (tid < CC) wl[tid] = wlin[tid];
    __syncthreads();
    int wave = tid >> 5, lane = tid & 31;
    int row = blockIdx.x * 8 + wave;
    float4 a = ((const float4*)(acc + (size_t)row * CC))[lane];
    float s = fmaxf(a.x, 0.f) * wl[lane * 4]     + fmaxf(a.y, 0.f) * wl[lane * 4 + 1] +
              fmaxf(a.z, 0.f) * wl[lane * 4 + 2] + fmaxf(a.w, 0.f) * wl[lane * 4 + 3];
    s = waveRedSum(s);
    if (lane == 0) out[row] = s + blin[0];
}

extern "C" void kernel_launch(void* const* d_in, const int* in_sizes, int n_in,
                              void* d_out, int out_size, void* d_ws, size_t ws_size,
                              hipStream_t stream) {
    const float* x      = (const float*)d_in[0];
    const int*   eidx   = (const int*)d_in[1];     // [2,E] row-major
    const float* ew     = (const float*)d_in[2];
    const float* pscore = (const float*)d_in[3];
    const float* W0     = (const float*)d_in[4];
    const float* w_ih   = (const float*)d_in[5];
    const float* w_hh   = (const float*)d_in[6];
    const float* b_ih   = (const float*)d_in[7];
    const float* b_hh   = (const float*)d_in[8];
    const float* w_lin  = (const float*)d_in[9];
    const float* b_lin  = (const float*)d_in[10];
    float* out = (float*)d_out;

    const int* erow = eidx;
    const int* ecol = eidx + EE;

    // workspace layout (floats)
    float* ws = (float*)d_ws;
    size_t o = 0;
    float* xw       = ws + o; o += (size_t)NN * CC;   // 6.4M
    float* acc      = ws + o; o += (size_t)NN * CC;   // 6.4M   (deg/dis follows contiguously)
    float* deg_dis  = ws + o; o += NN;
    float* score_f  = ws + o; o += NN;
    unsigned* score_u = (unsigned*)(ws + o); o += NN;
    float* xt       = ws + o; o += CC * CC;
    float* Wm       = ws + o; o += CC * CC;
    float* vals     = ws + o; o += 128;
    unsigned* perm  = (unsigned*)(ws + o); o += 128;
    unsigned* hist  = (unsigned*)(ws + o); o += 256;
    unsigned* st    = (unsigned*)(ws + o); o += 8;    // prefix, kremain, cand_cnt
    float* inv_norm = ws + o; o += 8;
    unsigned* cand_idx = (unsigned*)(ws + o); o += 1024;
    unsigned* cand_key = (unsigned*)(ws + o); o += 1024;

    // --- topk pipeline ---
    pnorm_kernel<<<1, 32, 0, stream>>>(pscore, inv_norm);
    score_kernel<<<NN / 8, 256, 0, stream>>>(x, pscore, inv_norm, score_f, score_u);
    rs_init<<<1, 256, 0, stream>>>(hist, st);
    for (int p = 0; p < 4; ++p) {
        rs_hist<<<512, 256, 0, stream>>>(score_u, hist, st, p);
        rs_scan<<<1, 1, 0, stream>>>(hist, st, p);
    }
    rs_collect<<<512, 256, 0, stream>>>(score_u, st, cand_idx, cand_key);
    rs_select<<<1, 128, 0, stream>>>(cand_idx, cand_key, st, score_f, perm, vals);
    build_xt<<<128, 128, 0, stream>>>(x, perm, vals, xt);

    // --- GRU -> evolved W ---
    gru_kernel<<<128, 384, 0, stream>>>(xt, W0, w_ih, w_hh, b_ih, b_hh, Wm);

    // --- zero acc + deg (contiguous region) ---
    zero_kernel<<<4096, 256, 0, stream>>>(acc, NN * CC + NN);

    // --- xw = x @ W (WMMA f32, async double-buffered tiles) ---
    gemm_xw<<<NN / (16 * TT), 256, 0, stream>>>(x, Wm, xw);

    // --- gcn_norm ---
    deg_kernel<<<2048, 256, 0, stream>>>(ecol, ew, deg_dis);
    rsqrt_kernel<<<(NN + 255) / 256, 256, 0, stream>>>(deg_dis);

    // --- scatter-sum over edges + self loops ---
    scatter_kernel<<<(EE + NN) / 8, 256, 0, stream>>>(erow, ecol, ew, deg_dis, xw, acc);

    // --- relu + head ---
    head_kernel<<<NN / 8, 256, 0, stream>>>(acc, w_lin, b_lin, out);
}